// DPS_topk_9088150798849
// MI455X (gfx1250) — compile-verified
//
#include <hip/hip_runtime.h>

#define VOCAB 2048
#define KSEL  16
#define TPB   256

typedef __attribute__((ext_vector_type(4))) float v4f;
// Match the builtin's parameter type exactly: 16-byte GCC vector of int.
typedef int v4i_vs __attribute__((vector_size(16)));
typedef __attribute__((address_space(1))) v4i_vs* gv4p;
typedef __attribute__((address_space(3))) v4i_vs* lv4p;
typedef __attribute__((address_space(3))) void*   as3vp;

// CDNA5 async global->LDS (ASYNCcnt path). Builtin when available, inline-asm otherwise.
static __device__ __forceinline__ void async_load_b128(const void* gsrc, void* ldst) {
#if __has_builtin(__builtin_amdgcn_global_load_async_to_lds_b128)
  __builtin_amdgcn_global_load_async_to_lds_b128((gv4p)gsrc, (lv4p)ldst, 0, 0);
#else
  unsigned loff = (unsigned)(unsigned long long)(as3vp)ldst;
  unsigned long long ga = (unsigned long long)gsrc;
  asm volatile("global_load_async_to_lds_b128 %0, %1, off" :: "v"(loff), "v"(ga) : "memory");
#endif
}

static __device__ __forceinline__ void wait_async0() {
#if __has_builtin(__builtin_amdgcn_s_wait_asynccnt)
  __builtin_amdgcn_s_wait_asynccnt(0);
#else
  asm volatile("s_wait_asynccnt 0x0" ::: "memory");
#endif
}

// POW2: d0 is a power of two -> d = blk & (d0-1), avoiding scalar integer division.
template <bool POW2>
__global__ __launch_bounds__(TPB) void dps_topk_kernel(
    const float* __restrict__ logits,
    const float* __restrict__ gn,
    float* __restrict__ out,
    int d0)
{
  __shared__ float s_logits[VOCAB];
  __shared__ float s_gn[VOCAB];
  __shared__ float s_work[VOCAB];
  __shared__ int   s_pick[KSEL];   // selection order (by value)
  __shared__ int   s_col[KSEL];    // per output row k: one-hot column (k-th smallest index)
  __shared__ float s_hard[KSEL];   // per output row k: value at the one-hot column

  const int tid = threadIdx.x;
  const int blk = blockIdx.x;              // = b*d0 + d
  const int d   = POW2 ? (blk & (d0 - 1)) : (blk - (blk / d0) * d0);

  const float* lrow = logits + (long)d   * VOCAB;
  const float* grow = gn     + (long)blk * VOCAB;

  // ---- stage logits row + gn row into LDS via async-to-LDS b128 (ASYNCcnt) ----
  #pragma unroll
  for (int j = 0; j < VOCAB / (TPB * 4); ++j) {
    const int c = (tid + TPB * j) * 4;     // first float of this 16B chunk
    async_load_b128(lrow + c, &s_logits[c]);
    async_load_b128(grow + c, &s_gn[c]);
  }
  wait_async0();
  __syncthreads();

  // perturbed row
  #pragma unroll
  for (int j = 0; j < VOCAB / TPB; ++j) {
    const int v = tid + TPB * j;
    s_work[v] = s_logits[v] + s_gn[v];
  }
  __syncthreads();

  // ---- wave 0: iterative top-16 argmax; per-lane register "taken" mask ----
  if (tid < 32) {
    unsigned long long taken = 0ull;       // bit j <=> element j*32+lane already selected
    for (int t = 0; t < KSEL; ++t) {
      float bv = -1.0e30f;
      int   bi = 0x7fffffff;
      #pragma unroll 8
      for (int j = 0; j < VOCAB / 32; ++j) {
        const int v = j * 32 + tid;        // bank-conflict-free
        const float x = ((taken >> j) & 1ull) ? -1.0e30f : s_work[v];
        if (x > bv || (x == bv && v < bi)) { bv = x; bi = v; }
      }
      #pragma unroll
      for (int off = 16; off > 0; off >>= 1) {  // butterfly: result in all lanes
        const float ov = __shfl_xor(bv, off, 32);
        const int   oi = __shfl_xor(bi, off, 32);
        if (ov > bv || (ov == bv && oi < bi)) { bv = ov; bi = oi; }
      }
      if ((bi & 31) == tid) taken |= 1ull << (bi >> 5);
      if (tid == 0) s_pick[t] = bi;
    }
  }
  __syncthreads();

  // ---- ascending rank of each selected index + hard-spot value ----
  if (tid < KSEL) {
    const int i = s_pick[tid];
    int p = 0;
    #pragma unroll
    for (int j = 0; j < KSEL; ++j) p += (s_pick[j] < i) ? 1 : 0;
    // at its own row the column is unmasked: soft = sigmoid(log(exp(l)*(1+EPS)) + g)
    const float l = s_logits[i], g = s_gn[i];
    const float x = logf(expf(l) * (1.0f + 1e-20f)) + g;
    const float s = 1.0f / (1.0f + expf(-x));
    s_col[p]  = i;
    s_hard[p] = (1.0f - s) + s;            // forward value of (hard - soft) + soft at hard==1
  }
  __syncthreads();

  // ---- stream the 16 x 2048 tile: zeros except one element per row ----
  // (hard - soft) + soft == 0.0f exactly wherever hard == 0, so only the one-hot
  // column per row is non-zero. 256 MB total output -> non-temporal b128 stores.
  v4f* ov4 = (v4f*)(out + (long)blk * (KSEL * VOCAB));
  #pragma unroll
  for (int j = 0; j < (KSEL * VOCAB / 4) / TPB; ++j) {
    const int e4 = tid + TPB * j;          // float4 index inside the tile
    const int k  = e4 >> 9;                // 512 float4 per row
    const int c4 = e4 & 511;
    const int ck = s_col[k];               // broadcast LDS read (uniform per pass)
    const bool hit = ((ck >> 2) == c4);
    const float h = hit ? s_hard[k] : 0.0f;
    const int comp = ck & 3;
    v4f val;
    val.x = (hit && comp == 0) ? h : 0.0f;
    val.y = (hit && comp == 1) ? h : 0.0f;
    val.z = (hit && comp == 2) ? h : 0.0f;
    val.w = (hit && comp == 3) ? h : 0.0f;
    __builtin_nontemporal_store(val, &ov4[e4]);
  }
}

extern "C" void kernel_launch(void* const* d_in, const int* in_sizes, int n_in,
                              void* d_out, int out_size, void* d_ws, size_t ws_size,
                              hipStream_t stream) {
  const float* logits = (const float*)d_in[0];
  const float* gn     = (const float*)d_in[1];
  float* out          = (float*)d_out;
  const int d0   = in_sizes[0] / VOCAB;    // 64
  const int rows = in_sizes[1] / VOCAB;    // BS * d0 = 2048 blocks
  if ((d0 & (d0 - 1)) == 0)
    dps_topk_kernel<true ><<<rows, TPB, 0, stream>>>(logits, gn, out, d0);
  else
    dps_topk_kernel<false><<<rows, TPB, 0, stream>>>(logits, gn, out, d0);
  (void)n_in; (void)out_size; (void)d_ws; (void)ws_size;
}